// SpikingCerebellum_88149908783723
// MI455X (gfx1250) — compile-verified
//
#include <hip/hip_runtime.h>
#include <hip/hip_bf16.h>

typedef __attribute__((ext_vector_type(2))) float v2f;
typedef __attribute__((ext_vector_type(8))) float v8f;

#define N_GC    8192
#define N_PC    2048
#define N_DCN   512
#define N_MOSSY 4096
#define N_IO    64
#define N_STEPS 20

// ---------------- workspace layout (floats) ----------------
enum : size_t {
  OFF_IGC   = 0,                     // raw W_mf_gc @ mossy     [N_GC]
  OFF_VG    = OFF_IGC  + N_GC,
  OFF_UG    = OFF_VG   + N_GC,
  OFF_RG    = OFF_UG   + N_GC,
  OFF_GCS   = OFF_RG   + N_GC,      // gc spike accumulator
  OFF_IPF   = OFF_GCS  + N_GC,      // 15*(W_pf @ rg)          [N_PC]
  OFF_ICF   = OFF_IPF  + N_PC,      // 5*(W_io_pc @ cf)        [N_PC]
  OFF_VP    = OFF_ICF  + N_PC,
  OFF_UP    = OFF_VP   + N_PC,
  OFF_RP    = OFF_UP   + N_PC,
  OFF_PCS   = OFF_RP   + N_PC,      // pc spike accumulator
  OFF_IDCN  = OFF_PCS  + N_PC,      // raw W_mf_dcn @ mossy    [N_DCN]
  OFF_VD    = OFF_IDCN + N_DCN,
  OFF_UD    = OFF_VD   + N_DCN,
  OFF_RD    = OFF_UD   + N_DCN,
  OFF_SUMRP = OFF_RD   + N_DCN,     // per-step sum(rp)        [N_STEPS]
  OFF_SCAL  = OFF_SUMRP + N_STEPS,  // [0]=sum|I_gc| [1]=sum|I_dcn| [2]=cf_scalar
  WS_FLOATS = OFF_SCAL + 4
};

// ---------------- init: state + accumulators ----------------
__global__ __launch_bounds__(256) void k_init(float* ws) {
  int i = blockIdx.x * 256 + threadIdx.x;
  if (i < N_GC)   { ws[OFF_VG+i] = -65.f; ws[OFF_UG+i] = -13.f; ws[OFF_RG+i] = 0.f; ws[OFF_GCS+i] = 0.f; }
  if (i < N_PC)   { ws[OFF_VP+i] = -65.f; ws[OFF_UP+i] = -13.f; ws[OFF_RP+i] = 0.f; ws[OFF_PCS+i] = 0.f; }
  if (i < N_DCN)  { ws[OFF_VD+i] = -65.f; ws[OFF_UD+i] = -13.f; ws[OFF_RD+i] = 0.f; }
  if (i < N_STEPS) ws[OFF_SUMRP+i] = 0.f;
  if (i < 4)       ws[OFF_SCAL+i]  = 0.f;
}

// ---------------- WMMA f32 matvec: y[16 rows per block] = scale * W @ x ----------------
// W row-major MxK. Block = 256 threads = 8 wave32 waves; each wave owns a K/8 slice.
// B = x-segment broadcast into all 16 columns so any D column = the 16 row partials.
// A 16x4 f32 layout (ISA 7.12.2): lane L in [0,16) holds row m0+L, K={0,1}+kb;
// lane L+16 holds row m0+L, K={2,3}+kb. B 4x16 mirrors the K mapping.
// 4 independent accumulators break the WMMA->WMMA RAW chain (4-way ILP); combined
// elementwise at the end (identical D layouts). K/8 is divisible by 16 for K=4096/8192.
__global__ __launch_bounds__(256) void k_wmma_matvec(
    const float* __restrict__ W, const float* __restrict__ x,
    float* __restrict__ y, int K, float scale, float* absAccum) {
  __shared__ float red[8 * 16];
  const int tid   = threadIdx.x;
  const int lane  = tid & 31;
  const int wave  = tid >> 5;
  const int m0    = blockIdx.x * 16;
  const int row   = m0 + (lane & 15);
  const int kHalf = (lane >> 4) << 1;     // 0 for lanes 0-15, 2 for lanes 16-31
  const int kPerWave = K >> 3;            // K / 8 waves
  const int kBeg  = wave * kPerWave;
  const int kEnd  = kBeg + kPerWave;

  const float* wRow = W + (size_t)row * (size_t)K;
  v8f acc0 = {}, acc1 = {}, acc2 = {}, acc3 = {};
  for (int kb = kBeg; kb < kEnd; kb += 16) {
    v2f a0 = *(const v2f*)(wRow + kb +  0 + kHalf);
    v2f b0 = *(const v2f*)(x    + kb +  0 + kHalf);
    v2f a1 = *(const v2f*)(wRow + kb +  4 + kHalf);
    v2f b1 = *(const v2f*)(x    + kb +  4 + kHalf);
    v2f a2 = *(const v2f*)(wRow + kb +  8 + kHalf);
    v2f b2 = *(const v2f*)(x    + kb +  8 + kHalf);
    v2f a3 = *(const v2f*)(wRow + kb + 12 + kHalf);
    v2f b3 = *(const v2f*)(x    + kb + 12 + kHalf);
    acc0 = __builtin_amdgcn_wmma_f32_16x16x4_f32(false, a0, false, b0, (short)0, acc0, false, false);
    acc1 = __builtin_amdgcn_wmma_f32_16x16x4_f32(false, a1, false, b1, (short)0, acc1, false, false);
    acc2 = __builtin_amdgcn_wmma_f32_16x16x4_f32(false, a2, false, b2, (short)0, acc2, false, false);
    acc3 = __builtin_amdgcn_wmma_f32_16x16x4_f32(false, a3, false, b3, (short)0, acc3, false, false);
  }
  v8f acc = (acc0 + acc1) + (acc2 + acc3);
  // D layout: lane l, VGPR r -> row m0 + r + 8*(l>>4), col l&15 (all cols equal).
  if ((lane & 15) == 0) {
    int base = wave * 16 + (lane >> 4) * 8;
#pragma unroll
    for (int r = 0; r < 8; ++r) red[base + r] = acc[r];
  }
  __syncthreads();
  if (tid < 16) {
    float s = 0.f;
#pragma unroll
    for (int w = 0; w < 8; ++w) s += red[w * 16 + tid];
    y[m0 + tid] = s * scale;
    if (absAccum) atomicAdd(absAccum, fabsf(s));   // for mean|I| normalization
  }
}

// ---------------- I_cf = 5 * (W_io_pc @ cf) ----------------
__global__ __launch_bounds__(256) void k_io(const float* __restrict__ Wio,
                                            const float* __restrict__ cf,
                                            float* __restrict__ Icf) {
  int i = blockIdx.x * 256 + threadIdx.x;
  if (i >= N_PC) return;
  const float* row = Wio + (size_t)i * N_IO;
  float s = 0.f;
#pragma unroll
  for (int k = 0; k < N_IO; ++k) s += row[k] * cf[k];
  Icf[i] = 5.0f * s;
}

// ---------------- cf_scalar = mean(cf) ----------------
__global__ void k_cfmean(const float* __restrict__ cf, float* out) {
  __shared__ float sm[64];
  int t = threadIdx.x;
  sm[t] = cf[t];
  __syncthreads();
  for (int s = 32; s > 0; s >>= 1) { if (t < s) sm[t] += sm[t + s]; __syncthreads(); }
  if (t == 0) out[0] = sm[0] * (1.0f / N_IO);
}

// ---------------- Izhikevich steps (reference order: v first, u uses NEW v) ------------
__global__ __launch_bounds__(256) void k_gc(const float* __restrict__ Iraw,
                                            const float* __restrict__ sumabs,
                                            const float* __restrict__ noise_t,
                                            float* v, float* u, float* r, float* sacc) {
  int i = blockIdx.x * 256 + threadIdx.x;
  float scale = 8.0f / (sumabs[0] * (1.0f / N_GC) + 1e-8f);
  float I = Iraw[i] * scale + noise_t[i] * 0.5f - 4.0f;       // i_tonic = -4
  float vv = v[i], uu = u[i];
  vv += 0.04f * vv * vv + 5.0f * vv + 140.0f - uu + I;
  uu += 0.02f * (0.2f * vv - uu);                             // RS: a=0.02 b=0.2
  float sp = (vv >= 30.0f) ? 1.0f : 0.0f;
  if (sp > 0.f) vv = -65.0f;                                  // c
  uu += sp * 8.0f;                                            // d
  r[i] = 0.95f * r[i] + 0.05f * sp;
  v[i] = vv; u[i] = uu;
  sacc[i] += sp;
}

__global__ __launch_bounds__(256) void k_pc(const float* __restrict__ Ipf,
                                            const float* __restrict__ Icf,
                                            float* v, float* u, float* r, float* sacc,
                                            float* sumrp_t) {
  __shared__ float sm[256];
  int i = blockIdx.x * 256 + threadIdx.x;
  float I = Ipf[i] + Icf[i] + 2.0f;                           // i_tonic = 2
  float vv = v[i], uu = u[i];
  vv += 0.04f * vv * vv + 5.0f * vv + 140.0f - uu + I;
  uu += 0.02f * (0.2f * vv - uu);                             // IB: a=0.02 b=0.2
  float sp = (vv >= 30.0f) ? 1.0f : 0.0f;
  if (sp > 0.f) vv = -55.0f;                                  // c
  uu += sp * 4.0f;                                            // d
  float rr = 0.95f * r[i] + 0.05f * sp;
  r[i] = rr; v[i] = vv; u[i] = uu; sacc[i] += sp;
  sm[threadIdx.x] = rr;
  __syncthreads();
  for (int s = 128; s > 0; s >>= 1) { if (threadIdx.x < s) sm[threadIdx.x] += sm[threadIdx.x + s]; __syncthreads(); }
  if (threadIdx.x == 0) atomicAdd(sumrp_t, sm[0]);            // sum(rp) for DCN drive
}

__global__ __launch_bounds__(256) void k_dcn(const float* __restrict__ Iraw,
                                             const float* __restrict__ sumabs,
                                             const float* __restrict__ sumrp_t,
                                             float* v, float* u, float* r) {
  int i = blockIdx.x * 256 + threadIdx.x;
  float scale = 3.0f / (sumabs[0] * (1.0f / N_DCN) + 1e-8f);
  // W_pc_dcn is the constant -0.5 matrix -> I_pc_dcn = 10*(-0.5*sum(rp)) = -5*sum(rp)
  float I = Iraw[i] * scale - 5.0f * sumrp_t[0] + 1.0f;       // i_tonic = 1
  float vv = v[i], uu = u[i];
  vv += 0.04f * vv * vv + 5.0f * vv + 140.0f - uu + I;
  uu += 0.02f * (0.2f * vv - uu);                             // RS
  float sp = (vv >= 30.0f) ? 1.0f : 0.0f;
  if (sp > 0.f) vv = -65.0f;
  uu += sp * 8.0f;
  r[i] = 0.95f * r[i] + 0.05f * sp;
  v[i] = vv; u[i] = uu;
}

// ---------------- small outputs: rd, gc_sparsity, pc_rate_mean ----------------
__global__ __launch_bounds__(1024) void k_finish(const float* __restrict__ ws,
                                                 float* __restrict__ out) {
  __shared__ float sm[1024];
  int t = threadIdx.x;
  float c = 0.f;
  for (int i = t; i < N_GC; i += 1024) c += (ws[OFF_GCS + i] > 0.f) ? 1.0f : 0.0f;
  sm[t] = c;
  __syncthreads();
  for (int s = 512; s > 0; s >>= 1) { if (t < s) sm[t] += sm[t + s]; __syncthreads(); }
  if (t < N_DCN) out[t] = ws[OFF_RD + t];
  if (t == 0) {
    out[N_DCN]     = sm[0] * (1.0f / N_GC);                       // gc_sparsity
    out[N_DCN + 1] = ws[OFF_SUMRP + N_STEPS - 1] * (1.0f / N_PC); // pc_rate_mean
  }
}

// ---------------- plasticity: elementwise over 2048x8192 (2 elems/thread) --------------
__global__ __launch_bounds__(256) void k_plastic(const float* __restrict__ Wpf,
                                                 const float* __restrict__ elig,
                                                 const float* __restrict__ gcs,
                                                 const float* __restrict__ pcs,
                                                 const float* __restrict__ cfp,
                                                 const float* __restrict__ DA,
                                                 float* __restrict__ Wout,
                                                 float* __restrict__ Eout) {
  size_t idx = ((size_t)blockIdx.x * 256 + threadIdx.x) * 2;
  int i = (int)(idx >> 13);            // row (PC), N_GC = 8192 = 2^13 cols
  int j = (int)(idx & (N_GC - 1));
  float cf = cfp[0];
  float pa  = (pcs[i]     > 0.f) ? 1.0f : 0.0f;
  float ga0 = (gcs[j]     > 0.f) ? 1.0f : 0.0f;
  float ga1 = (gcs[j + 1] > 0.f) ? 1.0f : 0.0f;
  v2f w = *(const v2f*)(Wpf  + idx);
  v2f e = *(const v2f*)(elig + idx);
  float el0 = 0.9f * e.x + pa * ga0;
  float el1 = 0.9f * e.y + pa * ga1;
  v2f wo, eo;
  if (cf > 0.1f) {
    float coef = 0.002f * cf * DA[0];
    eo.x = el0; eo.y = el1;
    wo.x = fminf(fmaxf(w.x - coef * el0, 0.01f), 1.0f);
    wo.y = fminf(fmaxf(w.y - coef * el1, 0.01f), 1.0f);
  } else {
    eo.x = 0.95f * e.x; eo.y = 0.95f * e.y;
    wo = w;
  }
  *(v2f*)(Wout + idx) = wo;
  *(v2f*)(Eout + idx) = eo;
}

// ---------------- host ----------------
extern "C" void kernel_launch(void* const* d_in, const int* in_sizes, int n_in,
                              void* d_out, int out_size, void* d_ws, size_t ws_size,
                              hipStream_t stream) {
  const float* mossy  = (const float*)d_in[0];   // [4096]
  const float* cf     = (const float*)d_in[1];   // [64]
  const float* DA     = (const float*)d_in[2];   // [1]
  const float* noise  = (const float*)d_in[3];   // [20,8192]
  const float* Wmfgc  = (const float*)d_in[4];   // [8192,4096]
  const float* Wpf    = (const float*)d_in[5];   // [2048,8192]
  /* d_in[6] = W_pc_dcn: constant -0.5 -> folded analytically */
  const float* Wmfdcn = (const float*)d_in[7];   // [512,4096]
  const float* Wio    = (const float*)d_in[8];   // [2048,64]
  const float* elig   = (const float*)d_in[9];   // [2048,8192]
  float* out = (float*)d_out;
  float* ws  = (float*)d_ws;

  float* Igc  = ws + OFF_IGC;  float* vg = ws + OFF_VG; float* ug = ws + OFF_UG;
  float* rg   = ws + OFF_RG;   float* gcs = ws + OFF_GCS;
  float* Ipf  = ws + OFF_IPF;  float* Icf = ws + OFF_ICF;
  float* vp = ws + OFF_VP; float* up = ws + OFF_UP; float* rp = ws + OFF_RP;
  float* pcs  = ws + OFF_PCS;  float* Idcn = ws + OFF_IDCN;
  float* vd = ws + OFF_VD; float* ud = ws + OFF_UD; float* rd = ws + OFF_RD;
  float* sumrp = ws + OFF_SUMRP; float* scal = ws + OFF_SCAL;

  k_init<<<N_GC / 256, 256, 0, stream>>>(ws);

  // one-shot input drives (WMMA matvecs, raw result + |.| accumulation)
  k_wmma_matvec<<<N_GC / 16, 256, 0, stream>>>(Wmfgc, mossy, Igc, N_MOSSY, 1.0f, scal + 0);
  k_wmma_matvec<<<N_DCN / 16, 256, 0, stream>>>(Wmfdcn, mossy, Idcn, N_MOSSY, 1.0f, scal + 1);
  k_io<<<N_PC / 256, 256, 0, stream>>>(Wio, cf, Icf);
  k_cfmean<<<1, 64, 0, stream>>>(cf, scal + 2);

  for (int t = 0; t < N_STEPS; ++t) {
    k_gc<<<N_GC / 256, 256, 0, stream>>>(Igc, scal + 0, noise + (size_t)t * N_GC,
                                         vg, ug, rg, gcs);
    k_wmma_matvec<<<N_PC / 16, 256, 0, stream>>>(Wpf, rg, Ipf, N_GC, 15.0f, nullptr);
    k_pc<<<N_PC / 256, 256, 0, stream>>>(Ipf, Icf, vp, up, rp, pcs, sumrp + t);
    k_dcn<<<N_DCN / 256, 256, 0, stream>>>(Idcn, scal + 1, sumrp + t, vd, ud, rd);
  }

  k_finish<<<1, 1024, 0, stream>>>(ws, out);

  const size_t nW = (size_t)N_PC * N_GC;              // 16777216
  k_plastic<<<(unsigned)(nW / 2 / 256), 256, 0, stream>>>(
      Wpf, elig, gcs, pcs, scal + 2, DA,
      out + N_DCN + 2, out + N_DCN + 2 + nW);
}